// SpectralStack_18940805775444
// MI455X (gfx1250) — compile-verified
//
#include <hip/hip_runtime.h>

// ---------------------------------------------------------------------------
// SpectralStack on MI455X (gfx1250), fp32 end-to-end via V_WMMA_F32_16X16X4_F32.
//
// times = repeat(arange(T), N) is already sorted -> stable argsort is the
// identity permutation -> the gather in _get_square_feats is a pure reshape.
//
// Per layer (per (b,t) slice, S = BS*T = 64 slices):
//   X1[c,e] = sum_n X[n,c] * eig[e,n]        (C x 2048)(2048 x 128) GEMM
//   X3[o,e] = sum_c X1[c,e] * p[o,c,e]       tiny, fused into k_front
//   Y[n,o]  = relu( sum_e X3[o,e]*eig[e,n] ) (O x 128)(128 x 2048) GEMM
//
// k_front streams the K=2048 reduction through LDS with REGISTER-STAGED
// DOUBLE BUFFERING: chunk i+1's global loads are in flight (LOADcnt) while
// chunk i is consumed by the WMMA loop, so L2/HBM latency is off the
// critical path even at 1 workgroup per WGP.
// ---------------------------------------------------------------------------

#define S_SLICES 64   // BS*T
#define NN       2048 // nodes per slice
#define EE       128  // eig dims
#define NCHUNK   128  // node chunk staged in LDS (k_front)
#define EG       32   // e-columns owned per workgroup (k_front)
#define NCHK2    128  // node chunk per workgroup (k_back)

typedef float v2f __attribute__((ext_vector_type(2)));
typedef float v8f __attribute__((ext_vector_type(8)));

// A (16x4, MxK) / B (4x16, KxN) fragment layout (ISA 7.12.2, 32-bit):
//   lane l: m/n = l%16,  K = 2*(l/16) + vgpr_index
// D (16x16 f32): vgpr v: lanes 0-15 -> (M=v, N=l); lanes 16-31 -> (M=v+8, N=l-16)

__device__ __forceinline__ v8f wmma_f32(v2f a, v2f b, v8f c) {
  return __builtin_amdgcn_wmma_f32_16x16x4_f32(false, a, false, b, (short)0, c,
                                               false, false);
}

template <int C, int O>
__global__ __launch_bounds__(256)
void k_front(const float* __restrict__ X,    // (S, NN, C)
             const float* __restrict__ eig,  // (EE, NN)
             const float* __restrict__ P,    // (O, C, EE)
             float* __restrict__ X3)         // (S, O, EE)
{
  constexpr int CROWS  = C / 16;             // 4 or 8
  constexpr int TPW    = (CROWS * 2) / 8;    // e-tiles per wave: 1 or 2
  constexpr int SE_STR = NCHUNK + 2;         // pad: break 16-way bank conflict
  constexpr int XR     = (NCHUNK * C) / 4 / 256;  // float4 regs staging X: 8|16
  constexpr int ER     = (EG * NCHUNK) / 2 / 256; // float2 regs staging eig: 8

  extern __shared__ float smem[];
  float* sX = smem;                          // [NCHUNK][C]   node-major chunk
  float* sE = smem + NCHUNK * C;             // [EG][SE_STR]  eig rows

  const int s    = blockIdx.x;
  const int e0   = blockIdx.y * EG;
  const int tid  = threadIdx.x;
  const int wave = tid >> 5;
  const int lane = tid & 31;
  const int m    = lane & 15;
  const int h    = lane >> 4;

  // Tile ownership: TPW==2 -> wave owns c-row `wave`, both e-tiles (shared A).
  //                 TPW==1 -> wave owns (c-row wave%CROWS, e-tile wave/CROWS).
  int c0, eBase;
  if constexpr (TPW == 2) { c0 = wave << 4;            eBase = 0; }
  else                    { c0 = (wave % CROWS) << 4;  eBase = (wave / CROWS) << 4; }

  const float* Xs = X + (size_t)s * NN * C;

  // ---- register staging buffers (double buffer front end) ----
  float4 xr[XR];
  float2 er[ER];

  auto load_chunk = [&](int n0) {            // global -> registers (in flight)
    const float4* src4 = (const float4*)(Xs + (size_t)n0 * C);
    #pragma unroll
    for (int i = 0; i < XR; ++i) xr[i] = src4[tid + i * 256];
    #pragma unroll
    for (int i = 0; i < ER; ++i) {
      const int idx = tid + i * 256;
      er[i] = *(const float2*)(eig + (size_t)(e0 + (idx >> 6)) * NN +
                               n0 + (idx & 63) * 2);
    }
  };
  auto store_chunk = [&]() {                 // registers -> LDS
    float4* dst4 = (float4*)sX;
    #pragma unroll
    for (int i = 0; i < XR; ++i) dst4[tid + i * 256] = xr[i];
    #pragma unroll
    for (int i = 0; i < ER; ++i) {
      const int idx = tid + i * 256;
      *(float2*)(sE + (idx >> 6) * SE_STR + (idx & 63) * 2) = er[i];
    }
  };

  v8f acc[TPW] = {};
  load_chunk(0);                             // prime the pipeline

  for (int n0 = 0; n0 < NN; n0 += NCHUNK) {
    store_chunk();                           // waits on in-flight loads only
    __syncthreads();
    if (n0 + NCHUNK < NN) load_chunk(n0 + NCHUNK);  // next chunk in flight

    #pragma unroll 4
    for (int k = 0; k < NCHUNK; k += 4) {
      const int kk = k + 2 * h;
      v2f a;
      a.x = sX[kk       * C + c0 + m];       // A[m, kk]   = X[n0+kk,   c0+m]
      a.y = sX[(kk + 1) * C + c0 + m];       // A[m, kk+1]
      #pragma unroll
      for (int j = 0; j < TPW; ++j) {
        const int eL = eBase + j * 16;
        const float2 bv = *(const float2*)(sE + (eL + m) * SE_STR + kk);
        v2f b; b.x = bv.x; b.y = bv.y;       // B[kk., m] = eig[e0+eL+m, n0+kk.]
        acc[j] = wmma_f32(a, b, acc[j]);
      }
    }
    __syncthreads();
  }

  // Spill X1 tiles to LDS (reuse sX region): sX1[c][e_local], C*EG floats
  float* sX1 = smem;
  #pragma unroll
  for (int j = 0; j < TPW; ++j) {
    const int eL = eBase + j * 16;
    #pragma unroll
    for (int v = 0; v < 8; ++v)
      sX1[(c0 + v + 8 * h) * EG + (eL + m)] = acc[j][v];
  }
  __syncthreads();

  // Fused filter contraction: X3[o, e0+e] = sum_c X1[c,e] * P[o,c,e0+e]
  #pragma unroll
  for (int it = 0; it < (O * EG) / 256; ++it) {
    const int idx = tid + it * 256;
    const int o = idx >> 5;                  // EG == 32
    const int e = idx & 31;
    const float* pp = P + (size_t)o * C * EE + e0 + e;
    float sum = 0.f;
    #pragma unroll 8
    for (int c = 0; c < C; ++c)
      sum = fmaf(sX1[c * EG + e], pp[(size_t)c * EE], sum);
    X3[((size_t)s * O + o) * EE + e0 + e] = sum;
  }
}

template <int O>
__global__ __launch_bounds__(256)
void k_back(const float* __restrict__ X3,    // (S, O, EE)
            const float* __restrict__ eig,   // (EE, NN)
            float* __restrict__ Y)           // (S, NN, O) = relu(X3 @ eig)^T
{
  constexpr int OROWS   = O / 16;            // 4 or 8
  constexpr int TPW     = OROWS;             // n-tiles per wave (one o-row each)
  constexpr int NSTRIDE = 8 / OROWS;         // n-tile stride between j's
  constexpr int SA_STR  = EE + 2;            // 130 (conflict pad, b64-aligned)
  constexpr int SB_STR  = NCHK2 + 4;         // 132 (conflict pad, b128-aligned)

  extern __shared__ float smem[];
  float* sA = smem;                          // [O][SA_STR]
  float* sB = smem + O * SA_STR;             // [EE][SB_STR]

  const int s    = blockIdx.x;
  const int n0   = blockIdx.y * NCHK2;
  const int tid  = threadIdx.x;
  const int wave = tid >> 5;
  const int lane = tid & 31;
  const int m    = lane & 15;
  const int h    = lane >> 4;

  // Stage A = X3 slice (contiguous O*EE) and B = eig panel (EE x NCHK2)
  {
    const float* A = X3 + (size_t)s * O * EE;
    #pragma unroll
    for (int i = 0; i < (O * EE) / 2 / 256; ++i) {
      const int idx = tid + i * 256;
      const int row = idx >> 6;              // EE/2 == 64
      const int c2  = idx & 63;
      *(float2*)(sA + row * SA_STR + c2 * 2) =
          *(const float2*)(A + row * EE + c2 * 2);
    }
    #pragma unroll
    for (int i = 0; i < (EE * NCHK2) / 4 / 256; ++i) {
      const int idx = tid + i * 256;
      const int row = idx >> 5;              // NCHK2/4 == 32
      const int c4  = idx & 31;
      *(float4*)(sB + row * SB_STR + c4 * 4) =
          *(const float4*)(eig + (size_t)row * NN + n0 + c4 * 4);
    }
  }
  __syncthreads();

  const int o0    = (wave % OROWS) << 4;     // wave's o-row (fixed across j)
  const int nbase = wave / OROWS;            // first n-tile

  v8f acc[TPW] = {};
  #pragma unroll 2
  for (int k = 0; k < EE; k += 4) {
    const int kk = k + 2 * h;
    const float2 av = *(const float2*)(sA + (o0 + m) * SA_STR + kk);
    v2f a; a.x = av.x; a.y = av.y;           // A[m, kk.] = X3[o0+m, kk.]
    #pragma unroll
    for (int j = 0; j < TPW; ++j) {
      const int nn0 = (nbase + j * NSTRIDE) << 4;
      v2f b;
      b.x = sB[kk       * SB_STR + nn0 + m]; // B[kk,   m] = eig[kk,   n0+nn0+m]
      b.y = sB[(kk + 1) * SB_STR + nn0 + m];
      acc[j] = wmma_f32(a, b, acc[j]);
    }
  }

  // D layout: lane -> node column (n = n0+nn0+m), vgpr -> o = o0 + 8h + v
  // => each lane owns 8 CONTIGUOUS output floats: relu + two b128 stores.
  #pragma unroll
  for (int j = 0; j < TPW; ++j) {
    const int nn0 = (nbase + j * NSTRIDE) << 4;
    float* yp = Y + ((size_t)s * NN + n0 + nn0 + m) * O + o0 + 8 * h;
    float4 r0, r1;
    r0.x = fmaxf(acc[j][0], 0.f); r0.y = fmaxf(acc[j][1], 0.f);
    r0.z = fmaxf(acc[j][2], 0.f); r0.w = fmaxf(acc[j][3], 0.f);
    r1.x = fmaxf(acc[j][4], 0.f); r1.y = fmaxf(acc[j][5], 0.f);
    r1.z = fmaxf(acc[j][6], 0.f); r1.w = fmaxf(acc[j][7], 0.f);
    *(float4*)(yp)     = r0;
    *(float4*)(yp + 4) = r1;
  }
}

extern "C" void kernel_launch(void* const* d_in, const int* in_sizes, int n_in,
                              void* d_out, int out_size, void* d_ws, size_t ws_size,
                              hipStream_t stream)
{
  (void)in_sizes; (void)n_in; (void)out_size; (void)ws_size;

  const float* feats = (const float*)d_in[0];
  // d_in[1] pos, d_in[2] times, d_in[3] ids: unused (gather is identity)
  const float* eig = (const float*)d_in[4];
  const float* p0  = (const float*)d_in[5];
  const float* p1  = (const float*)d_in[6];
  const float* p2  = (const float*)d_in[7];
  float* out = (float*)d_out;

  // Workspace: Y1 (64MB) | Y2 (64MB) | X3 scratch (4MB). Activations stay
  // L2-resident (192MB) between layers.
  float* Y1 = (float*)d_ws;
  float* Y2 = Y1 + (size_t)S_SLICES * NN * 128;
  float* X3 = Y2 + (size_t)S_SLICES * NN * 128;

  // k_back needs up to ~131KB dynamic LDS (WGP has 320KB)
  (void)hipFuncSetAttribute((const void*)(k_front<64, 128>),
      hipFuncAttributeMaxDynamicSharedMemorySize, 192 * 1024);
  (void)hipFuncSetAttribute((const void*)(k_front<128, 128>),
      hipFuncAttributeMaxDynamicSharedMemorySize, 192 * 1024);
  (void)hipFuncSetAttribute((const void*)(k_front<128, 64>),
      hipFuncAttributeMaxDynamicSharedMemorySize, 192 * 1024);
  (void)hipFuncSetAttribute((const void*)(k_back<128>),
      hipFuncAttributeMaxDynamicSharedMemorySize, 192 * 1024);
  (void)hipFuncSetAttribute((const void*)(k_back<64>),
      hipFuncAttributeMaxDynamicSharedMemorySize, 192 * 1024);

  const dim3 gF(S_SLICES, EE / EG);          // 64 x 4
  const dim3 gB(S_SLICES, NN / NCHK2);       // 64 x 16

  // Layer 0: C=64 -> O=128
  {
    constexpr size_t sm1 = (size_t)(NCHUNK * 64 + EG * (NCHUNK + 2)) * sizeof(float);
    constexpr size_t sm2 = (size_t)(128 * (EE + 2) + EE * (NCHK2 + 4)) * sizeof(float);
    k_front<64, 128><<<gF, 256, sm1, stream>>>(feats, eig, p0, X3);
    k_back<128><<<gB, 256, sm2, stream>>>(X3, eig, Y1);
  }
  // Layer 1: C=128 -> O=128
  {
    constexpr size_t sm1 = (size_t)(NCHUNK * 128 + EG * (NCHUNK + 2)) * sizeof(float);
    constexpr size_t sm2 = (size_t)(128 * (EE + 2) + EE * (NCHK2 + 4)) * sizeof(float);
    k_front<128, 128><<<gF, 256, sm1, stream>>>(Y1, eig, p1, X3);
    k_back<128><<<gB, 256, sm2, stream>>>(X3, eig, Y2);
  }
  // Layer 2: C=128 -> O=64
  {
    constexpr size_t sm1 = (size_t)(NCHUNK * 128 + EG * (NCHUNK + 2)) * sizeof(float);
    constexpr size_t sm2 = (size_t)(64 * (EE + 2) + EE * (NCHK2 + 4)) * sizeof(float);
    k_front<128, 64><<<gF, 256, sm1, stream>>>(Y2, eig, p2, X3);
    k_back<64><<<gB, 256, sm2, stream>>>(X3, eig, out);
  }
}